// CLAM_SB_27582279975346
// MI455X (gfx1250) — compile-verified
//
#include <hip/hip_runtime.h>
#include <hip/hip_bf16.h>

typedef float v2f __attribute__((ext_vector_type(2)));
typedef float v8f __attribute__((ext_vector_type(8)));

#define BB 8
#define NN 16384
#define DD 1024
#define AA 128
#define KS 64
#define NEG_INF_F (-1.0e30f)

// ---------------------------------------------------------------------------
// K1: f[b,n] = w2^T tanh(X W1 + b1) + b2, fused, WMMA f32 16x16x4.
// grid 512 blocks x 256 threads (8 waves); each wave computes 32 rows of f
// (two 16-row M tiles -> every B ds_load_b64 feeds two WMMAs).
// W1 staged through LDS in 64-row chunks, k-pairs interleaved so each lane's
// B operand is one contiguous 8B ds_load_b64 (no repack movs).
// LDS layout: w1s[((k>>1)*128 + n)*2 + (k&1)]
// ---------------------------------------------------------------------------
__global__ __launch_bounds__(256) void k_attn_f(
    const float* __restrict__ X, const float* __restrict__ W1,
    const float* __restrict__ b1, const float* __restrict__ w2,
    const float* __restrict__ b2, float* __restrict__ f) {
  __shared__ float w1s[64 * AA];  // 32 KB chunk of W1 (pair-interleaved)
  const int tid  = threadIdx.x;
  const int wave = tid >> 5;
  const int lane = tid & 31;
  const int lm = lane & 15;   // M index within tile (and N index for B/C)
  const int hi = lane >> 4;   // K-half selector
  const long rowBase = (long)blockIdx.x * 256 + wave * 32;
  const float* xrow0 = X + (rowBase + lm) * DD;
  const float* xrow1 = X + (rowBase + 16 + lm) * DD;

  v8f acc[16];
#pragma unroll
  for (int t = 0; t < 16; ++t)
#pragma unroll
    for (int e = 0; e < 8; ++e) acc[t][e] = 0.0f;

  for (int kc = 0; kc < DD; kc += 64) {
    __syncthreads();
    for (int j = tid; j < 64 * AA; j += 256) {
      int k = j >> 7, n = j & 127;
      w1s[((k >> 1) * AA + n) * 2 + (k & 1)] = W1[kc * AA + j];
    }
    __syncthreads();
#pragma unroll 4
    for (int kl = 0; kl < 64; kl += 4) {
      // A tiles 16x4 f32: lanes0-15 hold (K,K+1); lanes16-31 hold (K+2,K+3)
      v2f a0 = *(const v2f*)&xrow0[kc + kl + 2 * hi];
      v2f a1 = *(const v2f*)&xrow1[kc + kl + 2 * hi];
      const int bbase = (((kl >> 1) + hi) * AA + lm) * 2;
#pragma unroll
      for (int t = 0; t < 8; ++t) {
        v2f bb = *(const v2f*)&w1s[bbase + t * 32];
        acc[t] = __builtin_amdgcn_wmma_f32_16x16x4_f32(
            false, a0, false, bb, (short)0, acc[t], false, false);
        acc[8 + t] = __builtin_amdgcn_wmma_f32_16x16x4_f32(
            false, a1, false, bb, (short)0, acc[8 + t], false, false);
      }
    }
  }
  // Epilogue: h = tanh(acc + b1); f_row = sum_n h*w2; reduce 16 lanes/half.
#pragma unroll
  for (int half = 0; half < 2; ++half) {
    float partial[8];
#pragma unroll
    for (int r = 0; r < 8; ++r) partial[r] = 0.0f;
#pragma unroll
    for (int t = 0; t < 8; ++t) {
      const float b1v = b1[t * 16 + lm];
      const float w2v = w2[t * 16 + lm];
#pragma unroll
      for (int r = 0; r < 8; ++r)
        partial[r] += tanhf(acc[half * 8 + t][r] + b1v) * w2v;
    }
#pragma unroll
    for (int r = 0; r < 8; ++r) {
      float v = partial[r];
      v += __shfl_xor(v, 1, 32);
      v += __shfl_xor(v, 2, 32);
      v += __shfl_xor(v, 4, 32);
      v += __shfl_xor(v, 8, 32);
      if (lm == 0) f[rowBase + half * 16 + r + 8 * hi] = v + b2[0];
    }
  }
}

// ---------------------------------------------------------------------------
// K2: per-bag softmax stats (max, sumexp of masked f) + exact top-64 /
// bottom-64 index extraction (tie -> lower index, matches jax.lax.top_k).
// grid 8 x 256.
// ---------------------------------------------------------------------------
__global__ __launch_bounds__(256) void k_stats_topk(
    const float* __restrict__ f, const float* __restrict__ mask,
    float* __restrict__ fwork, float* __restrict__ stats,
    int* __restrict__ topIdx) {
  const int b = blockIdx.x, tid = threadIdx.x;
  const float* fb = f + b * NN;
  const float* mb = mask + b * NN;
  float* fw = fwork + b * NN;
  __shared__ float sv[256];
  __shared__ int   si[256];

  for (int n = tid; n < NN; n += 256) fw[n] = fb[n];

  float m = -3.0e38f;
  for (int n = tid; n < NN; n += 256) {
    float g = (mb[n] > 0.f) ? fb[n] : NEG_INF_F;
    m = fmaxf(m, g);
  }
  sv[tid] = m; __syncthreads();
  for (int s = 128; s > 0; s >>= 1) {
    if (tid < s) sv[tid] = fmaxf(sv[tid], sv[tid + s]);
    __syncthreads();
  }
  const float gmax = sv[0]; __syncthreads();
  float se = 0.f;
  for (int n = tid; n < NN; n += 256) {
    float g = (mb[n] > 0.f) ? fb[n] : NEG_INF_F;
    se += expf(g - gmax);
  }
  sv[tid] = se; __syncthreads();
  for (int s = 128; s > 0; s >>= 1) {
    if (tid < s) sv[tid] += sv[tid + s];
    __syncthreads();
  }
  if (tid == 0) { stats[b * 2] = gmax; stats[b * 2 + 1] = sv[0]; }
  __syncthreads();

  // top-64 (max extraction on raw f)
  for (int it = 0; it < KS; ++it) {
    float bv = -3.0e38f; int bidx = 0x3fffffff;
    for (int n = tid; n < NN; n += 256) {
      float v = fw[n];
      if (v > bv || (v == bv && n < bidx)) { bv = v; bidx = n; }
    }
    sv[tid] = bv; si[tid] = bidx; __syncthreads();
    for (int s = 128; s > 0; s >>= 1) {
      if (tid < s) {
        float ov = sv[tid + s]; int oi = si[tid + s];
        if (ov > sv[tid] || (ov == sv[tid] && oi < si[tid])) { sv[tid] = ov; si[tid] = oi; }
      }
      __syncthreads();
    }
    if (tid == 0) { topIdx[b * 128 + it] = si[0]; fw[si[0]] = -3.0e38f; }
    __syncthreads();
  }
  // restore extracted entries, then bottom-64 (min extraction)
  for (int i = tid; i < KS; i += 256) {
    int idx = topIdx[b * 128 + i];
    fw[idx] = fb[idx];
  }
  __syncthreads();
  for (int it = 0; it < KS; ++it) {
    float bv = 3.0e38f; int bidx = 0x3fffffff;
    for (int n = tid; n < NN; n += 256) {
      float v = fw[n];
      if (v < bv || (v == bv && n < bidx)) { bv = v; bidx = n; }
    }
    sv[tid] = bv; si[tid] = bidx; __syncthreads();
    for (int s = 128; s > 0; s >>= 1) {
      if (tid < s) {
        float ov = sv[tid + s]; int oi = si[tid + s];
        if (ov < sv[tid] || (ov == sv[tid] && oi < si[tid])) { sv[tid] = ov; si[tid] = oi; }
      }
      __syncthreads();
    }
    if (tid == 0) { topIdx[b * 128 + KS + it] = si[0]; fw[si[0]] = 3.0e38f; }
    __syncthreads();
  }
}

// ---------------------------------------------------------------------------
// K3: partial weighted sums z = sum_n softmax(f)_n * X[b,n,:].
// grid (8,16) x 256; block handles 1024 rows; thread owns 4 d-columns.
// ---------------------------------------------------------------------------
__global__ __launch_bounds__(256) void k_zpart(
    const float* __restrict__ X, const float* __restrict__ f,
    const float* __restrict__ mask, const float* __restrict__ stats,
    float* __restrict__ zpart) {
  const int b = blockIdx.x, c = blockIdx.y, tid = threadIdx.x;
  const float gmax = stats[b * 2];
  const float inv  = 1.0f / stats[b * 2 + 1];
  float a0 = 0.f, a1 = 0.f, a2 = 0.f, a3 = 0.f;
  const int n0 = c * 1024;
  for (int n = n0; n < n0 + 1024; ++n) {
    float g = (mask[b * NN + n] > 0.f) ? f[b * NN + n] : NEG_INF_F;
    float s = expf(g - gmax) * inv;
    const float* xr = X + ((long)(b * NN + n)) * DD;
    a0 += s * xr[tid];
    a1 += s * xr[tid + 256];
    a2 += s * xr[tid + 512];
    a3 += s * xr[tid + 768];
  }
  float* zp = zpart + (long)(b * 16 + c) * DD;
  zp[tid] = a0; zp[tid + 256] = a1; zp[tid + 512] = a2; zp[tid + 768] = a3;
}

// ---------------------------------------------------------------------------
// K4: z = fixed-order sum of 16 partials; bag_pred = z.Wc + bc. grid 8 x 256.
// ---------------------------------------------------------------------------
__global__ __launch_bounds__(256) void k_zreduce(
    const float* __restrict__ zpart, const float* __restrict__ Wc,
    const float* __restrict__ bc, float* __restrict__ bagpred,
    float* __restrict__ out) {
  const int b = blockIdx.x, tid = threadIdx.x;
  __shared__ float sv[256];
  float dot = 0.f;
  for (int j = 0; j < 4; ++j) {
    int d = tid + j * 256;
    float z = 0.f;
    for (int ch = 0; ch < 16; ++ch) z += zpart[(long)(b * 16 + ch) * DD + d];
    dot += z * Wc[d];
  }
  sv[tid] = dot; __syncthreads();
  for (int s = 128; s > 0; s >>= 1) {
    if (tid < s) sv[tid] += sv[tid + s];
    __syncthreads();
  }
  if (tid == 0) {
    float bp = sv[0] + bc[0];
    bagpred[b] = bp;
    out[b] = bp;
  }
}

// ---------------------------------------------------------------------------
// K5: instance logits + SmoothTop1SVM losses. grid 8 x 256.
// ---------------------------------------------------------------------------
__global__ __launch_bounds__(256) void k_inst(
    const float* __restrict__ X, const int* __restrict__ topIdx,
    const int* __restrict__ labels, const float* __restrict__ Wi,
    const float* __restrict__ bi, float* __restrict__ bagInst) {
  const int b = blockIdx.x, tid = threadIdx.x;
  const int lab = labels[b];
  __shared__ float lin[256];   // 128 instances x 2 classes (in-branch)
  __shared__ float lout[128];  // 64 x 2 (out-branch)
  __shared__ float red[256];
  {
    int i = tid >> 1, cls = tid & 1;
    int idx = topIdx[b * 128 + i];
    const float* xr = X + ((long)(b * NN + idx)) * DD;
    const float* w = Wi + (long)lab * DD * 2 + cls;  // Wi[lab][d][cls]
    float dp = 0.f;
    for (int d = 0; d < DD; ++d) dp += xr[d] * w[d * 2];
    lin[tid] = dp + bi[lab * 2 + cls];
  }
  if (tid < 128) {
    int i = tid >> 1, cls = tid & 1;     // i in [0,64): top_p rows
    int idx = topIdx[b * 128 + i];
    const float* xr = X + ((long)(b * NN + idx)) * DD;
    const float* w = Wi + (long)(1 - lab) * DD * 2 + cls;
    float dp = 0.f;
    for (int d = 0; d < DD; ++d) dp += xr[d] * w[d * 2];
    lout[tid] = dp + bi[(1 - lab) * 2 + cls];
  }
  __syncthreads();
  float contrib = 0.f;
  if (tid < 128) {  // loss_in, target: i<64 -> y=1 else y=0
    int y = (tid < KS) ? 1 : 0;
    float ly = lin[tid * 2 + y], lo = lin[tid * 2 + (1 - y)];
    float zo = lo + 1.0f - ly;                   // alpha=tau=1; z_y = 0
    float mm = fmaxf(0.f, zo);
    contrib += (mm + logf(expf(-mm) + expf(zo - mm))) * (1.0f / 128.0f);
  }
  if (tid < KS) {   // loss_out, y=0
    float ly = lout[tid * 2 + 0], lo = lout[tid * 2 + 1];
    float zo = lo + 1.0f - ly;
    float mm = fmaxf(0.f, zo);
    contrib += (mm + logf(expf(-mm) + expf(zo - mm))) * (1.0f / 64.0f);
  }
  red[tid] = contrib; __syncthreads();
  for (int s = 128; s > 0; s >>= 1) {
    if (tid < s) red[tid] += red[tid + s];
    __syncthreads();
  }
  if (tid == 0) bagInst[b] = red[0];
}

// ---------------------------------------------------------------------------
// K6: scalars: crit_loss (stable BCE-with-logits) + summed inst_loss.
// ---------------------------------------------------------------------------
__global__ void k_final(const float* __restrict__ bagpred,
                        const int* __restrict__ labels,
                        const float* __restrict__ bagInst,
                        float* __restrict__ out) {
  if (threadIdx.x == 0 && blockIdx.x == 0) {
    float crit = 0.f, inst = 0.f;
    for (int b = 0; b < BB; ++b) {
      float x = bagpred[b];
      float sp = fmaxf(x, 0.f) + log1pf(expf(-fabsf(x)));
      crit += sp - x * (float)labels[b];
      inst += bagInst[b];
    }
    out[8] = crit * (1.0f / (float)BB);
    out[9] = inst;
  }
}

extern "C" void kernel_launch(void* const* d_in, const int* in_sizes, int n_in,
                              void* d_out, int out_size, void* d_ws, size_t ws_size,
                              hipStream_t stream) {
  const float* X      = (const float*)d_in[0];
  const float* mask   = (const float*)d_in[1];
  const int*   labels = (const int*)  d_in[2];
  const float* W1     = (const float*)d_in[3];
  const float* b1     = (const float*)d_in[4];
  const float* w2     = (const float*)d_in[5];
  const float* b2     = (const float*)d_in[6];
  const float* Wc     = (const float*)d_in[7];
  const float* bc     = (const float*)d_in[8];
  const float* Wi     = (const float*)d_in[9];
  const float* bi     = (const float*)d_in[10];
  float* out = (float*)d_out;

  // Workspace layout (all rewritten every launch; no cross-call state).
  float* f       = (float*)d_ws;            // B*N            = 131072 f
  float* fwork   = f + BB * NN;             // B*N            = 131072 f
  float* stats   = fwork + BB * NN;         // B*2 (max, sumexp)
  int*   topIdx  = (int*)(stats + BB * 2);  // B*128 ints
  float* zpart   = (float*)(topIdx + BB * 128);  // B*16*D = 131072 f
  float* bagpred = zpart + BB * 16 * DD;    // B
  float* bagInst = bagpred + BB;            // B

  k_attn_f   <<<(BB * NN) / 256, 256, 0, stream>>>(X, W1, b1, w2, b2, f);
  k_stats_topk<<<BB, 256, 0, stream>>>(f, mask, fwork, stats, topIdx);
  k_zpart    <<<dim3(BB, 16), 256, 0, stream>>>(X, f, mask, stats, zpart);
  k_zreduce  <<<BB, 256, 0, stream>>>(zpart, Wc, bc, bagpred, out);
  k_inst     <<<BB, 256, 0, stream>>>(X, topIdx, labels, Wi, bi, bagInst);
  k_final    <<<1, 32, 0, stream>>>(bagpred, labels, bagInst, out);
}